// DepthRendSampler_77644418777273
// MI455X (gfx1250) — compile-verified
//
#include <hip/hip_runtime.h>
#include <hip/hip_bf16.h>
#include <stdint.h>

// ---------------- problem constants (from reference) ----------------
#define B_    4
#define ORD_  80
#define H_    176
#define W_    576
#define HW_   (H_ * W_)          // 101376 = 256*396, divisible by 16
#define C_    128
#define NSAMP 300
#define POOL_ 600                // top_k pool = 2*n_samp
#define NTOT  900
#define EPSF  1e-6f

typedef __attribute__((ext_vector_type(2))) float v2f;
typedef __attribute__((ext_vector_type(8))) float v8f;

// ---------------- CDNA5 feature detection (compile-safe) ----------------
#if defined(__AMDGCN__) && __has_builtin(__builtin_amdgcn_wmma_f32_16x16x4_f32)
#define HAVE_WMMA_F32X4 1
#else
#define HAVE_WMMA_F32X4 0
#endif

#if defined(__AMDGCN__) && __has_builtin(__builtin_amdgcn_global_load_async_to_lds_b32)
#define HAVE_ASYNC_LDS 1
#else
#define HAVE_ASYNC_LDS 0
#endif

__device__ __forceinline__ void wait_asynccnt0() {
#if defined(__AMDGCN__)
#if __has_builtin(__builtin_amdgcn_s_wait_asynccnt)
  __builtin_amdgcn_s_wait_asynccnt(0);
#else
  asm volatile("s_wait_asynccnt 0" ::: "memory");
#endif
#endif
}

__device__ __forceinline__ unsigned ballot32(int pred) {
#if defined(__AMDGCN__) && __has_builtin(__builtin_amdgcn_ballot_w32)
  return __builtin_amdgcn_ballot_w32(pred != 0);
#else
  return (unsigned)__ballot(pred);
#endif
}

// monotonic float -> uint key (larger key == larger float)
__device__ __forceinline__ unsigned fkey(float f) {
  unsigned u = __float_as_uint(f);
  return (u & 0x80000000u) ? ~u : (u | 0x80000000u);
}

// ==================================================================
// Kernel 1: entropy of ordinal distribution via V_WMMA_F32_16X16X4_F32
//   One wave32 computes 16 pixels. A(16x4) holds p*log(p) for 4 channels,
//   B(4x16) = ones, C/D f32 accumulates the channel sum over 20 WMMAs.
//   A layout (ISA 7.12.2): lanes 0-15 -> M=lane, VGPR0=K0, VGPR1=K1;
//                          lanes 16-31 -> M=lane-16, VGPR0=K2, VGPR1=K3.
//   D layout: VGPR j, lanes 0-15 -> D[j, lane]; lanes 16-31 -> D[8+j, lane-16].
//   All columns of D are equal (B is ones), so the diagonal gives entropy.
// ==================================================================
__global__ __launch_bounds__(256) void DRS_entropy_kernel(
    const float* __restrict__ p, float* __restrict__ ent) {
  int tid  = blockIdx.x * 256 + threadIdx.x;
  int wave = tid >> 5;
  int lane = tid & 31;
  int pix_base = wave << 4;                    // 16 pixels per wave
  int b   = pix_base / HW_;
  int pix = pix_base - b * HW_;
  int m   = lane & 15;
  const float* src = p + (size_t)b * ORD_ * HW_ + pix + m;

#if HAVE_WMMA_F32X4
  int kb = (lane >> 4) << 1;                   // 0 or 2
  v8f acc = {};
  v2f bones; bones.x = 1.0f; bones.y = 1.0f;
#pragma unroll 5
  for (int s = 0; s < ORD_ / 4; ++s) {
    int c0 = s * 4 + kb;
    float x0 = src[(size_t)c0 * HW_];
    float x1 = src[(size_t)(c0 + 1) * HW_];
    x0 = fminf(fmaxf(x0, EPSF), 1.0f - EPSF);
    x1 = fminf(fmaxf(x1, EPSF), 1.0f - EPSF);
    v2f a;
    a.x = x0 * __logf(x0);
    a.y = x1 * __logf(x1);
    acc = __builtin_amdgcn_wmma_f32_16x16x4_f32(
        false, a, false, bones, (short)0, acc, false, false);
  }
  float* dst = ent + (size_t)b * HW_ + pix;
  if (lane < 8) {
    dst[lane] = -acc[lane];                    // pixel m = lane (diagonal D[l,l])
  } else if (lane >= 16 && lane < 24) {
    dst[8 + (lane - 16)] = -acc[lane - 16];    // pixel 8+(lane-16)
  }
#else
  // scalar fallback (host pass / missing builtin)
  if (lane < 16) {
    float s = 0.0f;
    for (int ch = 0; ch < ORD_; ++ch) {
      float v = src[(size_t)ch * HW_];
      v = fminf(fmaxf(v, EPSF), 1.0f - EPSF);
      s += v * __logf(v);
    }
    ent[(size_t)b * HW_ + pix + m] = -s;
  }
#endif
}

// ==================================================================
// Kernel 2: 3x3 Laplace conv, replication padding.
//   18x18 halo tile staged in LDS via global_load_async_to_lds_b32.
//   conv weight = -laplace: corners -0.25, edges -0.5, center +3.0.
// ==================================================================
__global__ __launch_bounds__(256) void DRS_laplace_kernel(
    const float* __restrict__ ent, float* __restrict__ lap) {
  __shared__ float tile[18 * 19];
  int b  = blockIdx.z;
  int tx = threadIdx.x, ty = threadIdx.y;
  int x0 = blockIdx.x * 16, y0 = blockIdx.y * 16;
  const float* src = ent + (size_t)b * HW_;

  int t = ty * 16 + tx;
  for (int i = t; i < 18 * 18; i += 256) {
    int ly = i / 18, lx = i - ly * 18;
    int gy = min(max(y0 + ly - 1, 0), H_ - 1);
    int gx = min(max(x0 + lx - 1, 0), W_ - 1);
#if HAVE_ASYNC_LDS
    __builtin_amdgcn_global_load_async_to_lds_b32(
        (__attribute__((address_space(1))) int*)(src + (size_t)gy * W_ + gx),
        (__attribute__((address_space(3))) int*)&tile[ly * 19 + lx],
        0, 0);
#else
    tile[ly * 19 + lx] = src[(size_t)gy * W_ + gx];
#endif
  }
#if HAVE_ASYNC_LDS
  wait_asynccnt0();
#endif
  __syncthreads();

  const float* r0 = &tile[(ty + 0) * 19 + tx];
  const float* r1 = &tile[(ty + 1) * 19 + tx];
  const float* r2 = &tile[(ty + 2) * 19 + tx];
  float out = 3.0f * r1[1]
            - 0.5f  * (r0[1] + r1[0] + r1[2] + r2[1])
            - 0.25f * (r0[0] + r0[2] + r2[0] + r2[2]);
  lap[(size_t)b * HW_ + (size_t)(y0 + ty) * W_ + (x0 + tx)] = out;
}

// ==================================================================
// Kernel 3: per (batch, array) radix-select the key of the 600th-largest
//   element (MSB-first binary search over monotonic keys), plus the count
//   of elements strictly greater. One block per (b,arr); data is L2-resident.
// ==================================================================
__global__ __launch_bounds__(256) void DRS_select_kernel(
    const float* __restrict__ ent, const float* __restrict__ lap,
    unsigned* __restrict__ thresh, int* __restrict__ gcount) {
  int ba = blockIdx.x;                 // 0..7
  int b = ba >> 1, arr = ba & 1;
  const float* src = (arr ? lap : ent) + (size_t)b * HW_;
  __shared__ int s_cnt;
  int tid = threadIdx.x;

  unsigned prefix = 0;
  int remaining = POOL_;
  for (int bit = 31; bit >= 0; --bit) {
    unsigned cand   = prefix | (1u << bit);
    unsigned maskhi = ~((1u << bit) - 1u);
    if (tid == 0) s_cnt = 0;
    __syncthreads();
    int cnt = 0;
    for (int i = tid; i < HW_; i += 256)
      cnt += ((fkey(src[i]) & maskhi) == cand);
    atomicAdd(&s_cnt, cnt);
    __syncthreads();
    int c = s_cnt;
    if (c >= remaining) prefix = cand;
    else                remaining -= c;
    __syncthreads();                   // protect s_cnt reset next iteration
  }
  // count strictly greater than threshold
  if (tid == 0) s_cnt = 0;
  __syncthreads();
  int cg = 0;
  for (int i = tid; i < HW_; i += 256)
    cg += (fkey(src[i]) > prefix);
  atomicAdd(&s_cnt, cg);
  __syncthreads();
  if (tid == 0) { thresh[ba] = prefix; gcount[ba] = s_cnt; }
}

// ==================================================================
// Kernel 4: deterministic compaction of the top-600 pool in index order.
//   pool[0..G)   = indices with key >  T   (index-ascending)
//   pool[G..600) = first 600-G indices with key == T (index-ascending)
//   Ballot + LDS scan; no global atomics -> replay-deterministic.
// ==================================================================
__global__ __launch_bounds__(256) void DRS_compact_kernel(
    const float* __restrict__ ent, const float* __restrict__ lap,
    const unsigned* __restrict__ thresh, const int* __restrict__ gcount,
    int* __restrict__ pool) {
  int ba = blockIdx.x;
  int b = ba >> 1, arr = ba & 1;
  const float* src = (arr ? lap : ent) + (size_t)b * HW_;
  unsigned T = thresh[ba];
  int G = gcount[ba];
  int* mypool = pool + ba * POOL_;

  __shared__ int s_gt_wave[8], s_eq_wave[8];
  __shared__ int s_gt_base, s_eq_base;
  int tid = threadIdx.x, wave = tid >> 5, lane = tid & 31;
  if (tid == 0) { s_gt_base = 0; s_eq_base = 0; }
  __syncthreads();

  for (int base = 0; base < HW_; base += 256) {   // HW_ % 256 == 0
    int i = base + tid;
    unsigned k = fkey(src[i]);
    int gt = (k > T), eq = (k == T);
    unsigned mg = ballot32(gt);
    unsigned me = ballot32(eq);
    if (lane == 0) { s_gt_wave[wave] = __popc(mg); s_eq_wave[wave] = __popc(me); }
    __syncthreads();
    int goff = s_gt_base, eoff = s_eq_base;
    for (int w = 0; w < wave; ++w) { goff += s_gt_wave[w]; eoff += s_eq_wave[w]; }
    unsigned lt = (1u << lane) - 1u;
    int pg = goff + __popc(mg & lt);
    int pe = G + eoff + __popc(me & lt);
    if (gt) mypool[pg] = i;
    if (eq && pe < POOL_) mypool[pe] = i;
    __syncthreads();
    if (tid == 0) {
      int tg = 0, te = 0;
      for (int w = 0; w < 8; ++w) { tg += s_gt_wave[w]; te += s_eq_wave[w]; }
      s_gt_base += tg; s_eq_base += te;
    }
    __syncthreads();
  }
}

// ==================================================================
// Kernel 5: assemble idx_flat[b][900]:
//   [0,300)   : subset-300 of ent pool via coprime-stride permutation
//   [300,600) : subset-300 of lap pool via a different permutation
//   [600,900) : 300 distinct pseudo-random pixels (stride coprime to HW)
//   Also emit idx_hw and idx_flat outputs (value-cast to float).
// ==================================================================
__global__ __launch_bounds__(256) void DRS_build_idx_kernel(
    const int* __restrict__ pool, int* __restrict__ idx,
    float* __restrict__ out_idxhw, float* __restrict__ out_idxflat) {
  int t = blockIdx.x * 256 + threadIdx.x;
  if (t >= B_ * NTOT) return;
  int b = t / NTOT, j = t - b * NTOT;
  int id;
  if (j < NSAMP) {
    unsigned pj = ((unsigned)j * 343u + 17u) % (unsigned)POOL_;   // gcd(343,600)=1
    id = pool[(b * 2 + 0) * POOL_ + (int)pj];
  } else if (j < 2 * NSAMP) {
    unsigned jj = (unsigned)(j - NSAMP);
    unsigned pj = (jj * 491u + 89u) % (unsigned)POOL_;            // gcd(491,600)=1
    id = pool[(b * 2 + 1) * POOL_ + (int)pj];
  } else {
    unsigned jj = (unsigned)(j - 2 * NSAMP);
    id = (int)((jj * 48611u + (unsigned)b * 7919u + 12345u) % (unsigned)HW_); // 48611 coprime to HW
  }
  idx[t] = id;
  out_idxhw[(size_t)t * 2 + 0] = (float)(id / W_);
  out_idxhw[(size_t)t * 2 + 1] = (float)(id % W_);
  out_idxflat[t] = (float)id;
}

// ==================================================================
// Kernel 6: gather sampled features: out[b][c][0][j] = feat[b][c][idx[b][j]]
// ==================================================================
__global__ __launch_bounds__(256) void DRS_gather_kernel(
    const float* __restrict__ feat, const int* __restrict__ idx,
    float* __restrict__ out) {
  int t = blockIdx.x * 256 + threadIdx.x;
  if (t >= B_ * C_ * NTOT) return;
  int j  = t % NTOT;
  int bc = t / NTOT;                 // b*C + c
  int b  = bc / C_;
  out[t] = feat[(size_t)bc * HW_ + idx[b * NTOT + j]];
}

// ==================================================================
extern "C" void kernel_launch(void* const* d_in, const int* in_sizes, int n_in,
                              void* d_out, int out_size, void* d_ws, size_t ws_size,
                              hipStream_t stream) {
  (void)in_sizes; (void)n_in; (void)out_size; (void)ws_size;
  const float* p_bin = (const float*)d_in[0];   // [4,80,176,576]
  const float* feat  = (const float*)d_in[1];   // [4,128,176,576]
  float* out = (float*)d_out;

  // workspace layout (~3.3 MB)
  float*    ent    = (float*)d_ws;                       // B*HW
  float*    lap    = ent + (size_t)B_ * HW_;             // B*HW
  unsigned* thresh = (unsigned*)(lap + (size_t)B_ * HW_);// 8
  int*      gcount = (int*)(thresh + 8);                 // 8
  int*      pool   = gcount + 8;                         // 8*600
  int*      idx    = pool + 8 * POOL_;                   // B*900

  // output layout: sampled [4,128,1,900] | idx_hw [4,1,900,2] | idx_flat [4,1,900]
  float* out_sampled = out;
  float* out_idxhw   = out + (size_t)B_ * C_ * NTOT;             // +460800
  float* out_idxflat = out_idxhw + (size_t)B_ * NTOT * 2;        // +7200

  // 1) entropy: B*HW/16 waves -> 811008 threads -> 3168 blocks of 256
  DRS_entropy_kernel<<<(B_ * HW_ / 16) * 32 / 256, 256, 0, stream>>>(p_bin, ent);

  // 2) laplace conv with async-LDS tile staging
  dim3 lgrid(W_ / 16, H_ / 16, B_);
  DRS_laplace_kernel<<<lgrid, dim3(16, 16), 0, stream>>>(ent, lap);

  // 3) threshold select + 4) compaction: one block per (batch, array)
  DRS_select_kernel<<<2 * B_, 256, 0, stream>>>(ent, lap, thresh, gcount);
  DRS_compact_kernel<<<2 * B_, 256, 0, stream>>>(ent, lap, thresh, gcount, pool);

  // 5) build indices + emit idx outputs
  DRS_build_idx_kernel<<<(B_ * NTOT + 255) / 256, 256, 0, stream>>>(
      pool, idx, out_idxhw, out_idxflat);

  // 6) gather sampled features
  DRS_gather_kernel<<<(B_ * C_ * NTOT + 255) / 256, 256, 0, stream>>>(
      feat, idx, out_sampled);
}